// Loss_31903017074985
// MI455X (gfx1250) — compile-verified
//
#include <hip/hip_runtime.h>
#include <hip/hip_bf16.h>
#include <math.h>

typedef __attribute__((ext_vector_type(2))) float v2f;
typedef __attribute__((ext_vector_type(8))) float v8f;

// ---------------------------------------------------------------------------
// Workspace layout (floats):
// [0]=mask_sum [1]=den [2]=num [3]=normal_sum [4]=bce_sum
// [5]=eik_sum  [6]=sdf_sum [7]=con_sum [8]=vis_total [9]=depth_l1 [10]=pc_sum
// ---------------------------------------------------------------------------

__device__ __forceinline__ float nan0f(float x) {
    return isfinite(x) ? x : 0.0f;
}

__device__ __forceinline__ float waveSum32(float v) {
    #pragma unroll
    for (int off = 16; off; off >>= 1) v += __shfl_xor(v, off, 32);
    return v;
}

__device__ __forceinline__ void waveAtomicAdd(float* dst, float v) {
    v = waveSum32(v);
    if ((threadIdx.x & 31) == 0) atomicAdd(dst, v);
}

// ---------------------------------------------------------------------------
// Pass 1 over H*W pixels: mask_sum, depth den/num, normal L2 sum, BCE sum
// ---------------------------------------------------------------------------
__global__ void loss_pass1_kernel(const float* __restrict__ normal_pred,
                                  const float* __restrict__ normal_gt,
                                  const float* __restrict__ depth_pred,
                                  const float* __restrict__ depth_gt,
                                  const float* __restrict__ mask_raw,
                                  const float* __restrict__ comp_mask,
                                  int HW, float* __restrict__ ws) {
    float msum = 0.f, den = 0.f, num = 0.f, nsum = 0.f, bce = 0.f;
    for (int i = blockIdx.x * blockDim.x + threadIdx.x; i < HW;
         i += gridDim.x * blockDim.x) {
        const float m = (mask_raw[i] > 0.5f) ? 1.0f : 0.0f;
        msum += m;

        // depth pass-1: den = sum(vg*vg), num = sum(vg*vp) over valid pixels
        float g = (m == 1.0f) ? depth_gt[i] : 0.0f;
        float p = (m == 1.0f) ? depth_pred[i] : 0.0f;
        g = nan0f(g); p = nan0f(p);
        den += g * g;
        num += g * p;

        // masked normal L2
        #pragma unroll
        for (int c = 0; c < 3; ++c) {
            const float a = nan0f(normal_pred[3 * i + c]);
            const float b = nan0f(normal_gt[3 * i + c]);
            const float e = (a - b) * m;
            nsum += e * e;
        }

        // BCE on comp_mask vs binarized mask
        float cm = comp_mask[i];
        if (isnan(cm)) cm = 0.5f;
        cm = fminf(fmaxf(cm, 0.0f), 1.0f);
        cm = fminf(fmaxf(cm, 1e-5f), 1.0f - 1e-5f);
        bce += -(m * logf(cm) + (1.0f - m) * logf(1.0f - cm));
    }
    waveAtomicAdd(&ws[0], msum);
    waveAtomicAdd(&ws[1], den);
    waveAtomicAdd(&ws[2], num);
    waveAtomicAdd(&ws[3], nsum);
    waveAtomicAdd(&ws[4], bce);
}

// ---------------------------------------------------------------------------
// Eikonal + SDF L1 over N samples
// ---------------------------------------------------------------------------
__global__ void eik_sdf_kernel(const float* __restrict__ gradients,
                               const float* __restrict__ sdf,
                               int N, float* __restrict__ ws) {
    float eik = 0.f, sdfs = 0.f;
    for (int i = blockIdx.x * blockDim.x + threadIdx.x; i < N;
         i += gridDim.x * blockDim.x) {
        const float gx = gradients[3 * i + 0];
        const float gy = gradients[3 * i + 1];
        const float gz = gradients[3 * i + 2];
        const float gn = sqrtf(gx * gx + gy * gy + gz * gz) - 1.0f;
        eik += gn * gn;
        sdfs += fabsf(sdf[i]);
    }
    waveAtomicAdd(&ws[5], eik);
    waveAtomicAdd(&ws[6], sdfs);
}

// ---------------------------------------------------------------------------
// Consistency loss over P points, C cameras
// ---------------------------------------------------------------------------
__global__ void consistency_kernel(const float* __restrict__ nw,
                                   const float* __restrict__ vis,
                                   const float* __restrict__ w,
                                   int P, int C, float* __restrict__ ws) {
    float con = 0.f, vtot = 0.f;
    for (int p = blockIdx.x * blockDim.x + threadIdx.x; p < P;
         p += gridDim.x * blockDim.x) {
        const float wp = w[p];
        for (int i = 0; i < C - 1; ++i) {
            float mse = 0.f;
            #pragma unroll
            for (int k = 0; k < 3; ++k) {
                const float d = nw[(p * C + i) * 3 + k] - nw[(p * C + i + 1) * 3 + k];
                mse += d * d;
            }
            const float v = vis[p * C + i] * vis[p * C + i + 1];
            vtot += v;
            con += mse * v * wp;
        }
    }
    waveAtomicAdd(&ws[7], con);
    waveAtomicAdd(&ws[8], vtot);
}

// ---------------------------------------------------------------------------
// Depth pass-2: L1 of (vg*scale - vp); scale derived from pass-1 sums in ws
// ---------------------------------------------------------------------------
__global__ void depth_pass2_kernel(const float* __restrict__ depth_pred,
                                   const float* __restrict__ depth_gt,
                                   const float* __restrict__ mask_raw,
                                   int HW, float* __restrict__ ws) {
    float scale = ws[2] / ws[1];                  // num/den
    if (!isfinite(scale)) scale = 1.0f;           // nan_to_num(nan=1, +-inf=1)
    float l1 = 0.f;
    for (int i = blockIdx.x * blockDim.x + threadIdx.x; i < HW;
         i += gridDim.x * blockDim.x) {
        const float m = (mask_raw[i] > 0.5f) ? 1.0f : 0.0f;
        float g = (m == 1.0f) ? depth_gt[i] : 0.0f;
        float p = (m == 1.0f) ? depth_pred[i] : 0.0f;
        g = nan0f(g); p = nan0f(p);
        l1 += fabsf(g * scale - p);
    }
    waveAtomicAdd(&ws[9], l1);
}

// ---------------------------------------------------------------------------
// Chamfer (one direction per launch) via WMMA:
//   d2[m][n] = |x_m|^2 + |y_n|^2 - 2 x_m.y_n  computed as D = A x B + C with
//   A = x tile (16x4 f32, K=3 + zero pad), B = -2*y^T (4x16),
//   C[m][n] = |x_m|^2 + |y_n|^2.
// One wave owns a 16-source tile; the block stages destination points through
// LDS in 2048-point chunks as float4(x,y,z,|y|^2) so the hot loop is
// ds_load_b128 -> v_wmma_f32_16x16x4_f32 -> min/argmin update.
// ---------------------------------------------------------------------------
#define CHAMFER_CHUNK 2048   // points per LDS chunk: 2048 * 16B = 32 KB

__global__ void chamfer_wmma_kernel(const float* __restrict__ src,
                                    const float* __restrict__ dst,
                                    int S, float* __restrict__ acc) {
    __shared__ float4 buf[CHAMFER_CHUNK];

    const int lane = threadIdx.x & 31;
    const int half = lane >> 4;                   // 0: M rows 0..7, 1: rows 8..15
    const int j = lane & 15;
    const int tile = blockIdx.x * (blockDim.x >> 5) + (threadIdx.x >> 5);
    const int m0 = tile * 16;

    // ---- A matrix (16x4 f32): lanes 0-15 hold (K0,K1), lanes 16-31 (K2,0) ----
    int mp = m0 + j;
    if (mp >= S) mp = S - 1;                      // clamp (keeps barriers uniform)
    const float sx0 = src[3 * mp + 0];
    const float sx1 = src[3 * mp + 1];
    const float sx2 = src[3 * mp + 2];
    v2f a;
    a.x = half ? sx2 : sx0;
    a.y = half ? 0.0f : sx1;

    // |x_m|^2 for the 8 M-rows this half owns (M = r + 8*half)
    const float snorm = sx0 * sx0 + sx1 * sx1 + sx2 * sx2;
    float xn[8];
    #pragma unroll
    for (int r = 0; r < 8; ++r)
        xn[r] = __shfl(snorm, r + 8 * half, 32);

    float minv[8];
    int mini[8];
    #pragma unroll
    for (int r = 0; r < 8; ++r) { minv[r] = 3.4e38f; mini[r] = 0; }

    for (int base = 0; base < S; base += CHAMFER_CHUNK) {
        __syncthreads();
        // cooperative fill: (x,y,z,|y|^2); pad OOB with huge norm so it never wins
        for (int i = threadIdx.x; i < CHAMFER_CHUNK; i += blockDim.x) {
            const int gi = base + i;
            float4 v;
            if (gi < S) {
                v.x = dst[3 * gi + 0];
                v.y = dst[3 * gi + 1];
                v.z = dst[3 * gi + 2];
                v.w = v.x * v.x + v.y * v.y + v.z * v.z;
            } else {
                v.x = 0.f; v.y = 0.f; v.z = 0.f; v.w = 1.0e37f;
            }
            buf[i] = v;
        }
        __syncthreads();

        #pragma unroll 2
        for (int t = 0; t < CHAMFER_CHUNK / 16; ++t) {
            const int nl = t * 16 + j;            // this lane's column point
            const float4 y = buf[nl];             // ds_load_b128
            // B matrix (4x16): lanes 0-15 rows (K0,K1), lanes 16-31 (K2,0)
            v2f b;
            b.x = half ? (-2.0f * y.z) : (-2.0f * y.x);
            b.y = half ? 0.0f : (-2.0f * y.y);
            v8f c;
            #pragma unroll
            for (int r = 0; r < 8; ++r) c[r] = xn[r] + y.w;

            v8f d = __builtin_amdgcn_wmma_f32_16x16x4_f32(
                false, a, false, b, (short)0, c, false, false);

            const int n = base + nl;
            #pragma unroll
            for (int r = 0; r < 8; ++r) {
                const float dv = d[r];
                if (dv < minv[r]) { minv[r] = dv; mini[r] = n; }
            }
        }
    }

    // Min-reduce (value, index) over the 16 lanes of each half (same M rows).
    #pragma unroll
    for (int r = 0; r < 8; ++r) {
        float v = minv[r];
        int idx = mini[r];
        #pragma unroll
        for (int off = 8; off; off >>= 1) {
            const float ov = __shfl_xor(v, off, 32);
            const int oi = __shfl_xor(idx, off, 32);
            if (ov < v || (ov == v && oi < idx)) { v = ov; idx = oi; }
        }
        mini[r] = idx;
    }

    // Lanes j==0 of each half finalize their 8 source points with the exact
    // gather-then-norm of the reference (vec = x - y[idx]).
    if (j == 0 && m0 < S) {
        float lsum = 0.f;
        #pragma unroll
        for (int r = 0; r < 8; ++r) {
            const int m = m0 + r + 8 * half;
            if (m < S) {
                const int idx = mini[r];
                const float dx = src[3 * m + 0] - dst[3 * idx + 0];
                const float dy = src[3 * m + 1] - dst[3 * idx + 1];
                const float dz = src[3 * m + 2] - dst[3 * idx + 2];
                lsum += sqrtf(dx * dx + dy * dy + dz * dz);
            }
        }
        atomicAdd(acc, lsum);
    }
}

// ---------------------------------------------------------------------------
// Finalize: combine all terms with reference epsilons / nan handling
// ---------------------------------------------------------------------------
__global__ void finalize_kernel(const float* __restrict__ ws,
                                float* __restrict__ out,
                                int S, int HW, int N) {
    const float msum = ws[0], den = ws[1];
    const float mask_sum = msum + 1e-5f;

    const float normal_loss = ws[3] / mask_sum;

    float depth_loss = ws[9] / (mask_sum + 1e-8f);
    if (den < 1e-10f) depth_loss = 0.0f;
    depth_loss = nan0f(depth_loss);

    const float pc_loss = ws[10] / (float)S;      // both directions, /S each
    const float mask_loss = ws[4] / (float)HW;
    const float eik_loss = ws[5] / (float)N;
    const float sdf_loss = ws[6] / (float)N;
    const float con_loss = (ws[8] > 0.0f) ? (ws[7] / ws[8]) : 0.0f;

    out[0] = normal_loss + depth_loss + pc_loss + mask_loss + eik_loss +
             sdf_loss + con_loss;
}

// ---------------------------------------------------------------------------
extern "C" void kernel_launch(void* const* d_in, const int* in_sizes, int n_in,
                              void* d_out, int out_size, void* d_ws, size_t ws_size,
                              hipStream_t stream) {
    const float* normal_pred = (const float*)d_in[0];
    const float* normal_gt   = (const float*)d_in[1];
    const float* depth_pred  = (const float*)d_in[2];
    const float* depth_gt    = (const float*)d_in[3];
    const float* X           = (const float*)d_in[4];
    const float* Y           = (const float*)d_in[5];
    const float* mask_raw    = (const float*)d_in[6];
    const float* comp_mask   = (const float*)d_in[7];
    const float* gradients   = (const float*)d_in[8];
    const float* sdf         = (const float*)d_in[9];
    const float* nw_all      = (const float*)d_in[10];
    const float* vis_mask    = (const float*)d_in[11];
    const float* weights     = (const float*)d_in[12];

    const int HW = in_sizes[2];         // depth_pred: H*W
    const int S  = in_sizes[4] / 3;     // X: (1,S,3)
    const int N  = in_sizes[9];         // sdf: (N,)
    const int P  = in_sizes[12];        // weights: (P,)
    const int C  = in_sizes[11] / P;    // visibility: (P,C)

    float* ws = (float*)d_ws;
    hipMemsetAsync(d_ws, 0, 16 * sizeof(float), stream);

    loss_pass1_kernel<<<512, 256, 0, stream>>>(
        normal_pred, normal_gt, depth_pred, depth_gt, mask_raw, comp_mask, HW, ws);

    eik_sdf_kernel<<<256, 256, 0, stream>>>(gradients, sdf, N, ws);

    consistency_kernel<<<128, 256, 0, stream>>>(nw_all, vis_mask, weights, P, C, ws);

    // One launch per chamfer direction; one wave per 16-source tile, 8/block.
    const int tiles = (S + 15) / 16;
    const int chamferBlocks = (tiles + 7) / 8;
    chamfer_wmma_kernel<<<chamferBlocks, 256, 0, stream>>>(X, Y, S, &ws[10]);
    chamfer_wmma_kernel<<<chamferBlocks, 256, 0, stream>>>(Y, X, S, &ws[10]);

    depth_pass2_kernel<<<512, 256, 0, stream>>>(
        depth_pred, depth_gt, mask_raw, HW, ws);

    finalize_kernel<<<1, 1, 0, stream>>>(ws, (float*)d_out, S, HW, N);
}